// CoordAtt_52312701665891
// MI455X (gfx1250) — compile-verified
//
#include <hip/hip_runtime.h>
#include <cmath>

#define EPSF 1e-5f
#define NB 8
#define CCH 256
#define HH 112
#define WW 112
#define MIP 8
#define LL 112

// workspace layout (float offsets)
#define OFF_XMEAN 0                       // [N][2][C][L]  side0 = h-means, side1 = w-means
#define SZ_XMEAN  (NB*2*CCH*LL)
#define OFF_ATT   (OFF_XMEAN + SZ_XMEAN)  // [N][2][C][L]  sigmoid attention maps
#define SZ_ATT    (NB*2*CCH*LL)
#define OFF_DBUF  (OFF_ATT + SZ_ATT)      // [N][2][16][L] relu'd down-conv (K padded to 16)
#define SZ_DBUF   (NB*2*16*LL)
#define OFF_EFFW  (OFF_DBUF + SZ_DBUF)    // [2][16][C]    BN-folded down weights (M padded)
#define SZ_EFFW   (2*16*CCH)
#define OFF_EFFB  (OFF_EFFW + SZ_EFFW)    // [2][16]
#define SZ_EFFB   (2*16)
#define OFF_USWP  (OFF_EFFB + SZ_EFFB)    // [C][16]       up weights, K padded 8->16
#define SZ_USWP   (CCH*16)

typedef float v2f __attribute__((ext_vector_type(2)));
typedef float v4f __attribute__((ext_vector_type(4)));
typedef float v8f __attribute__((ext_vector_type(8)));

__device__ __forceinline__ v8f wmma_f32_k4(v2f a, v2f b, v8f c) {
  // V_WMMA_F32_16X16X4_F32 : D(16x16,f32) = A(16x4,f32) * B(4x16,f32) + C
  return __builtin_amdgcn_wmma_f32_16x16x4_f32(false, a, false, b, (short)0, c, false, false);
}

// ---------------------------------------------------------------------------
// Kernel P: fold BN into down-conv weights; pad matrices for 16x16 WMMA tiles.
// ---------------------------------------------------------------------------
__global__ void coordatt_prep(const float* __restrict__ g1, const float* __restrict__ b1,
                              const float* __restrict__ m1, const float* __restrict__ v1,
                              const float* __restrict__ g2, const float* __restrict__ b2,
                              const float* __restrict__ m2, const float* __restrict__ v2,
                              const float* __restrict__ dsw, const float* __restrict__ dsb,
                              const float* __restrict__ usw, float* __restrict__ ws) {
  __shared__ float s_scale[2][CCH];
  __shared__ float s_shift[2][CCH];
  int c = threadIdx.x;  // 256 threads
  float sc1 = g1[c] * rsqrtf(v1[c] + EPSF);
  s_scale[0][c] = sc1; s_shift[0][c] = b1[c] - m1[c] * sc1;
  float sc2 = g2[c] * rsqrtf(v2[c] + EPSF);
  s_scale[1][c] = sc2; s_shift[1][c] = b2[c] - m2[c] * sc2;
  __syncthreads();

  float* effw = ws + OFF_EFFW;
  float* effb = ws + OFF_EFFB;
  float* uswp = ws + OFF_USWP;

  #pragma unroll
  for (int side = 0; side < 2; ++side)
    for (int m = 0; m < 16; ++m)
      effw[(side*16 + m)*CCH + c] = (m < MIP) ? dsw[m*CCH + c] * s_scale[side][c] : 0.0f;

  for (int i = threadIdx.x; i < CCH*16; i += blockDim.x) {
    int cc = i >> 4, k = i & 15;
    uswp[i] = (k < MIP) ? usw[cc*MIP + k] : 0.0f;
  }

  if (threadIdx.x < 32) {
    int side = threadIdx.x >> 4, m = threadIdx.x & 15;
    float s = 0.0f;
    if (m < MIP) {
      s = dsb[m];
      for (int cc = 0; cc < CCH; ++cc) s += dsw[m*CCH + cc] * s_shift[side][cc];
    }
    effb[side*16 + m] = s;
  }
}

// ---------------------------------------------------------------------------
// Kernel A: dual-axis mean pooling. One block per (n,c) 112x112 slab.
// ---------------------------------------------------------------------------
__global__ void coordatt_pool(const float* __restrict__ x, float* __restrict__ ws) {
  int nc = blockIdx.x;                    // [0, N*C)
  int n = nc / CCH, c = nc % CCH;
  const float* xp = x + (size_t)nc * (HH*WW);
  int wave = threadIdx.x >> 5, lane = threadIdx.x & 31;
  float* xh = ws + OFF_XMEAN + ((size_t)(n*2 + 0)*CCH + c)*LL;
  float* xw = ws + OFF_XMEAN + ((size_t)(n*2 + 1)*CCH + c)*LL;

  __shared__ float colp[8][WW];
  float c0 = 0.f, c1 = 0.f, c2 = 0.f, c3 = 0.f;
  for (int h = wave; h < HH; h += 8) {
    const float* row = xp + h*WW;
    float v0 = row[lane];
    float v1 = row[lane + 32];
    float v2 = row[lane + 64];
    float v3 = (lane < 16) ? row[lane + 96] : 0.0f;
    c0 += v0; c1 += v1; c2 += v2; c3 += v3;
    float rs = v0 + v1 + v2 + v3;            // row (over-W) partial
    #pragma unroll
    for (int off = 16; off > 0; off >>= 1) rs += __shfl_down(rs, off, 32);
    if (lane == 0) xh[h] = rs * (1.0f/112.0f);
  }
  colp[wave][lane]      = c0;
  colp[wave][lane + 32] = c1;
  colp[wave][lane + 64] = c2;
  if (lane < 16) colp[wave][lane + 96] = c3;
  __syncthreads();
  for (int w = threadIdx.x; w < WW; w += blockDim.x) {
    float s = 0.f;
    #pragma unroll
    for (int q = 0; q < 8; ++q) s += colp[q][w];
    xw[w] = s * (1.0f/112.0f);               // column (over-H) mean
  }
}

// ---------------------------------------------------------------------------
// Kernel B: attention chain via V_WMMA_F32_16X16X4_F32.
// One block per (n, side). Down: M=16(pad) N=112 K=256. Up: M=256 N=112 K=16.
// Fragment layouts per ISA 7.12.2:
//   A (16x4): lane l -> M=l%16, VGPR v -> K = 4k + 2*(l/16) + v
//   B (4x16): lane l -> N=l%16, VGPR v -> K = 4k + 2*(l/16) + v
//   C/D     : lane l, VGPR r -> M = r + 8*(l/16), N = l%16
// ---------------------------------------------------------------------------
__global__ void coordatt_att(const float* __restrict__ usb, float* __restrict__ ws) {
  int n = blockIdx.x >> 1, side = blockIdx.x & 1;
  int wave = threadIdx.x >> 5, lane = threadIdx.x & 31;
  int lm = lane & 15;
  int kh = lane >> 4;                        // 0 or 1: K sub-pair select

  const float* xm = ws + OFF_XMEAN + (size_t)(n*2 + side)*CCH*LL;   // [C][L]
  const float* ew = ws + OFF_EFFW + side*16*CCH;                     // [16][C]
  const float* eb = ws + OFF_EFFB + side*16;                         // [16]
  const float* uw = ws + OFF_USWP;                                   // [C][16]
  float* dp = ws + OFF_DBUF + (size_t)(n*2 + side)*16*LL;            // [16][L]
  float* ap = ws + OFF_ATT  + (size_t)(n*2 + side)*CCH*LL;           // [C][L]

  // ---- down conv (BN folded) + ReLU : 7 N-tiles on waves 0..6
  if (wave < 7) {
    int ncol = wave*16 + lm;
    v8f acc = {0,0,0,0,0,0,0,0};
    for (int k = 0; k < CCH/4; ++k) {
      int k0 = 4*k + 2*kh;
      v2f A; A.x = ew[lm*CCH + k0];        A.y = ew[lm*CCH + k0 + 1];
      v2f B; B.x = xm[k0*LL + ncol];       B.y = xm[(k0+1)*LL + ncol];
      acc = wmma_f32_k4(A, B, acc);
    }
    #pragma unroll
    for (int r = 0; r < 8; ++r) {
      int m = r + 8*kh;
      float v = (m < MIP) ? fmaxf(acc[r] + eb[m], 0.0f) : 0.0f;  // padded rows stay 0
      dp[m*LL + ncol] = v;
    }
  }
  __syncthreads();

  // ---- up conv + sigmoid : 16 M-tiles x 7 N-tiles = 112 tiles over 8 waves
  for (int t = wave; t < 112; t += 8) {
    int mt = t / 7, nt = t % 7;
    int ncol = nt*16 + lm;
    v8f acc = {0,0,0,0,0,0,0,0};
    #pragma unroll
    for (int k = 0; k < 4; ++k) {          // K = 16 (padded)
      int k0 = 4*k + 2*kh;
      v2f A; A.x = uw[(mt*16 + lm)*16 + k0]; A.y = uw[(mt*16 + lm)*16 + k0 + 1];
      v2f B; B.x = dp[k0*LL + ncol];         B.y = dp[(k0+1)*LL + ncol];
      acc = wmma_f32_k4(A, B, acc);
    }
    #pragma unroll
    for (int r = 0; r < 8; ++r) {
      int m = mt*16 + r + 8*kh;
      float v = acc[r] + usb[m];
      ap[m*LL + ncol] = 1.0f / (1.0f + __expf(-v));
    }
  }
}

// ---------------------------------------------------------------------------
// Kernel C: elementwise apply + h_swish, one float4-worth per thread (exact
// tiling: N*C*H*(W/4) = 25088 * 256). Nontemporal b128 stores for the output.
// ---------------------------------------------------------------------------
__global__ void coordatt_apply(const float* __restrict__ x, const float* __restrict__ ws,
                               float* __restrict__ out) {
  int idx = blockIdx.x * blockDim.x + threadIdx.x;   // v4f index
  int w4 = idx % (WW/4);
  int t  = idx / (WW/4);
  int h  = t % HH;
  int nc = t / HH;
  int n = nc >> 8, c = nc & 255;

  const float* att = ws + OFF_ATT;
  float ah = att[((size_t)(n*2 + 0)*CCH + c)*LL + h];
  v4f aw = *reinterpret_cast<const v4f*>(att + ((size_t)(n*2 + 1)*CCH + c)*LL + w4*4);
  v4f xv = reinterpret_cast<const v4f*>(x)[idx];

  v4f o;
  #pragma unroll
  for (int i = 0; i < 4; ++i) {
    float v  = xv[i]*aw[i]*ah + xv[i];
    float r6 = fminf(fmaxf(v + 3.0f, 0.0f), 6.0f);
    o[i] = v * r6 * (1.0f/6.0f);
  }
  __builtin_nontemporal_store(o, reinterpret_cast<v4f*>(out) + idx);
}

// ---------------------------------------------------------------------------
extern "C" void kernel_launch(void* const* d_in, const int* in_sizes, int n_in,
                              void* d_out, int out_size, void* d_ws, size_t ws_size,
                              hipStream_t stream) {
  (void)in_sizes; (void)n_in; (void)out_size; (void)ws_size;
  const float* x   = (const float*)d_in[0];
  const float* g1  = (const float*)d_in[1];
  const float* b1  = (const float*)d_in[2];
  const float* m1  = (const float*)d_in[3];
  const float* v1  = (const float*)d_in[4];
  const float* g2  = (const float*)d_in[5];
  const float* b2  = (const float*)d_in[6];
  const float* m2  = (const float*)d_in[7];
  const float* v2  = (const float*)d_in[8];
  const float* dsw = (const float*)d_in[9];
  const float* dsb = (const float*)d_in[10];
  const float* usw = (const float*)d_in[11];
  const float* usb = (const float*)d_in[12];
  float* ws  = (float*)d_ws;
  float* out = (float*)d_out;

  coordatt_prep <<<1,        256, 0, stream>>>(g1,b1,m1,v1,g2,b2,m2,v2,dsw,dsb,usw,ws);
  coordatt_pool <<<NB*CCH,   256, 0, stream>>>(x, ws);
  coordatt_att  <<<NB*2,     256, 0, stream>>>(usb, ws);
  coordatt_apply<<<(NB*CCH*HH*(WW/4))/256, 256, 0, stream>>>(x, ws, out);
}